// DotProductAttention_26268019982930
// MI455X (gfx1250) — compile-verified
//
#include <hip/hip_runtime.h>

// ---------------------------------------------------------------------------
// out = (Q K^T) V  ==  Q (K^T V)   (no softmax -> associativity holds)
// B=2,H=8 -> 16 heads; S=4096; D=64. All fp32, WMMA f32 16x16x4.
// ---------------------------------------------------------------------------

typedef float v2f __attribute__((ext_vector_type(2)));
typedef float v8f __attribute__((ext_vector_type(8)));

#define ATT_S   4096
#define ATT_D   64
#define NHEAD   16                   // B*H
#define NCHUNK  16                   // S-chunks per head in stage 1
#define CHUNK_S (ATT_S / NCHUNK)     // 256 rows per workgroup
#define K1_WAVES 4
#define WAVE_S  (CHUNK_S / K1_WAVES) // 64 rows per wave
#define QTILE   128                  // q rows per workgroup in stage 3
#define QWAVES  2
#define QPAD    68                   // LDS row pad: %4==0 (float4), %64!=0 (banks)

static __device__ __forceinline__ v8f wmma_f32(v2f a, v2f b, v8f c) {
  // D = A(16x4,f32) * B(4x16,f32) + C(16x16,f32)
  return __builtin_amdgcn_wmma_f32_16x16x4_f32(false, a, false, b,
                                               (short)0, c, false, false);
}

// ---------------------------------------------------------------------------
// Stage 1: partial[head][chunk] = K_chunk^T * V_chunk  (64x64 f32)
// A[d1,k] = K[s0+k, d1], B[k,d2] = V[s0+k, d2]; both loads are 16 consecutive
// floats across lanes 0..15 -> coalesced 64B segments, no LDS staging needed.
// ---------------------------------------------------------------------------
__global__ __launch_bounds__(K1_WAVES * 32)
void attn_ktv_partial(const float* __restrict__ K, const float* __restrict__ V,
                      float* __restrict__ partial) {
  const int head  = blockIdx.x / NCHUNK;
  const int chunk = blockIdx.x % NCHUNK;
  const int wave  = threadIdx.x >> 5;
  const int lane  = threadIdx.x & 31;
  const int lmod  = lane & 15;
  const int lhalf = lane >> 4;

  const float* __restrict__ Kh = K + (size_t)head * ATT_S * ATT_D;
  const float* __restrict__ Vh = V + (size_t)head * ATT_S * ATT_D;
  const int s0 = chunk * CHUNK_S + wave * WAVE_S;

  v8f c[4][4];
#pragma unroll
  for (int i = 0; i < 4; ++i)
#pragma unroll
    for (int j = 0; j < 4; ++j)
      c[i][j] = (v8f){0.f, 0.f, 0.f, 0.f, 0.f, 0.f, 0.f, 0.f};

#pragma unroll 2
  for (int step = 0; step < WAVE_S / 4; ++step) {
    const int s = s0 + step * 4 + lhalf * 2;   // this lane-half's K rows s, s+1
    const float* __restrict__ kr = Kh + (size_t)s * ATT_D + lmod;
    const float* __restrict__ vr = Vh + (size_t)s * ATT_D + lmod;
    v2f a[4], b[4];
#pragma unroll
    for (int t = 0; t < 4; ++t) {
      a[t].x = kr[t * 16];            // A tile t: rows d1 = t*16 + lmod
      a[t].y = kr[t * 16 + ATT_D];
      b[t].x = vr[t * 16];            // B tile t: cols d2 = t*16 + lmod
      b[t].y = vr[t * 16 + ATT_D];
    }
#pragma unroll
    for (int i = 0; i < 4; ++i)
#pragma unroll
      for (int j = 0; j < 4; ++j)
        c[i][j] = wmma_f32(a[i], b[j], c[i][j]);
  }

  // cross-wave reduction through LDS (fixed order -> deterministic)
  __shared__ float red[K1_WAVES][ATT_D * ATT_D];   // 64 KB
#pragma unroll
  for (int i = 0; i < 4; ++i)
#pragma unroll
    for (int j = 0; j < 4; ++j)
#pragma unroll
      for (int r = 0; r < 8; ++r)
        red[wave][(i * 16 + r + lhalf * 8) * ATT_D + j * 16 + lmod] = c[i][j][r];
  __syncthreads();

  float* __restrict__ outp =
      partial + ((size_t)head * NCHUNK + chunk) * (ATT_D * ATT_D);
  for (int e = threadIdx.x; e < ATT_D * ATT_D; e += K1_WAVES * 32)
    outp[e] = (red[0][e] + red[1][e]) + (red[2][e] + red[3][e]);
}

// ---------------------------------------------------------------------------
// Stage 2: M[head] = sum over chunks of partial[head][chunk]  (fixed order)
// ---------------------------------------------------------------------------
__global__ __launch_bounds__(256)
void attn_reduce(const float* __restrict__ partial, float* __restrict__ M) {
  const int head = blockIdx.x;
  const float* __restrict__ p = partial + (size_t)head * NCHUNK * ATT_D * ATT_D;
  for (int e = threadIdx.x; e < ATT_D * ATT_D; e += 256) {
    float s = 0.f;
#pragma unroll
    for (int cc = 0; cc < NCHUNK; ++cc)
      s += p[cc * ATT_D * ATT_D + e];
    M[(size_t)head * ATT_D * ATT_D + e] = s;
  }
}

// ---------------------------------------------------------------------------
// Stage 3: out[head][q][:] = Q[head][q][:] @ M[head]
// Q staged through padded LDS (coalesced float4 fill; conflict-free f32x2
// gathers for the WMMA A tiles). M read from global (L2/WGP$-resident).
// ---------------------------------------------------------------------------
__global__ __launch_bounds__(QWAVES * 32)
void attn_qm(const float* __restrict__ Q, const float* __restrict__ M,
             float* __restrict__ out) {
  const int head  = blockIdx.x / (ATT_S / QTILE);
  const int qc    = blockIdx.x % (ATT_S / QTILE);
  const int wave  = threadIdx.x >> 5;
  const int lane  = threadIdx.x & 31;
  const int lmod  = lane & 15;
  const int lhalf = lane >> 4;
  const int q0    = qc * QTILE;

  __shared__ float qs[QTILE * QPAD];   // ~34.8 KB

  const float* __restrict__ Qh = Q + ((size_t)head * ATT_S + q0) * ATT_D;
  for (int idx = threadIdx.x; idx < QTILE * (ATT_D / 4); idx += QWAVES * 32) {
    const int row = idx >> 4;          // 16 float4 per 64-float row
    const int c4  = idx & 15;
    const float4 f = ((const float4*)Qh)[idx];
    *(float4*)&qs[row * QPAD + c4 * 4] = f;
  }
  __syncthreads();

  const float* __restrict__ Mh = M + (size_t)head * ATT_D * ATT_D;

  v8f c[4][4];
#pragma unroll
  for (int i = 0; i < 4; ++i)
#pragma unroll
    for (int j = 0; j < 4; ++j)
      c[i][j] = (v8f){0.f, 0.f, 0.f, 0.f, 0.f, 0.f, 0.f, 0.f};

#pragma unroll 2
  for (int step = 0; step < ATT_D / 4; ++step) {
    const int k0 = step * 4 + lhalf * 2;       // contraction rows k0, k0+1
    v2f a[4], b[4];
#pragma unroll
    for (int t = 0; t < 4; ++t) {
      // A tile t: rows q = wave*64 + t*16 + lmod ; 8B-aligned LDS read
      const v2f av = *(const v2f*)&qs[(wave * 64 + t * 16 + lmod) * QPAD + k0];
      a[t] = av;
      // B tile t: cols d2 = t*16 + lmod
      b[t].x = Mh[(k0 + 0) * ATT_D + t * 16 + lmod];
      b[t].y = Mh[(k0 + 1) * ATT_D + t * 16 + lmod];
    }
#pragma unroll
    for (int i = 0; i < 4; ++i)
#pragma unroll
      for (int j = 0; j < 4; ++j)
        c[i][j] = wmma_f32(a[i], b[j], c[i][j]);
  }

  float* __restrict__ Oh =
      out + ((size_t)head * ATT_S + q0 + wave * 64) * ATT_D;
#pragma unroll
  for (int i = 0; i < 4; ++i)
#pragma unroll
    for (int j = 0; j < 4; ++j)
#pragma unroll
      for (int r = 0; r < 8; ++r)
        Oh[(size_t)(i * 16 + r + lhalf * 8) * ATT_D + j * 16 + lmod] =
            c[i][j][r];
}

// ---------------------------------------------------------------------------
extern "C" void kernel_launch(void* const* d_in, const int* in_sizes, int n_in,
                              void* d_out, int out_size, void* d_ws,
                              size_t ws_size, hipStream_t stream) {
  (void)in_sizes; (void)n_in; (void)out_size; (void)ws_size;
  const float* q = (const float*)d_in[0];
  const float* k = (const float*)d_in[1];
  const float* v = (const float*)d_in[2];
  float* out = (float*)d_out;

  float* partial = (float*)d_ws;                                   // 4 MB
  float* M = partial + (size_t)NHEAD * NCHUNK * ATT_D * ATT_D;     // 256 KB

  attn_ktv_partial<<<NHEAD * NCHUNK, K1_WAVES * 32, 0, stream>>>(k, v, partial);
  attn_reduce<<<NHEAD, 256, 0, stream>>>(partial, M);
  attn_qm<<<NHEAD * (ATT_S / QTILE), QWAVES * 32, 0, stream>>>(q, M, out);
}